// MoleculeEncoder_39539468927577
// MI455X (gfx1250) — compile-verified
//
#include <hip/hip_runtime.h>

typedef float v2f __attribute__((ext_vector_type(2)));
typedef float v8f __attribute__((ext_vector_type(8)));

#define DFEAT 32

// ---------------------------------------------------------------------------
// WMMA fp32 GEMM:  out[nRows, outC] = A[nRows, DFEAT] @ W[DFEAT, outC] (+bias)
// One wave computes one 16x16 output tile; K=32 via 8 x V_WMMA_F32_16X16X4_F32.
// A-matrix VGPR layout (16x4 f32): lanes 0-15 -> M=lane, K={k,k+1};
//                                  lanes 16-31 -> M=lane-16, K={k+2,k+3}.
// B-matrix (4x16) symmetric; C/D: VGPR v -> M=v (+8 for upper half), N=lane&15.
// ---------------------------------------------------------------------------
__global__ void gemm_k32_wmma(const float* __restrict__ A,
                              const float* __restrict__ W,
                              const float* __restrict__ bias,
                              float* __restrict__ out,
                              int nRows, int outC) {
  const int colTiles = outC >> 4;
  const int totalTiles = (nRows >> 4) * colTiles;
  const int wave = blockIdx.x * (blockDim.x >> 5) + (threadIdx.x >> 5);
  if (wave >= totalTiles) return;  // wave-uniform branch; EXEC all-1 inside

  const int rowTile = wave / colTiles;
  const int colTile = wave - rowTile * colTiles;
  const int lane = threadIdx.x & 31;
  const int m = lane & 15;
  const int half = lane >> 4;  // 0: K+{0,1}, 1: K+{2,3}
  const int row0 = rowTile << 4;
  const int col0 = colTile << 4;

  const float* __restrict__ arow = A + (size_t)(row0 + m) * DFEAT;

  v8f acc = {};
#pragma unroll
  for (int k0 = 0; k0 < DFEAT; k0 += 4) {
    const int ka = k0 + (half << 1);
    v2f a, b;
    a.x = arow[ka];
    a.y = arow[ka + 1];
    b.x = W[(size_t)(ka + 0) * outC + col0 + m];
    b.y = W[(size_t)(ka + 1) * outC + col0 + m];
    acc = __builtin_amdgcn_wmma_f32_16x16x4_f32(false, a, false, b,
                                                (short)0, acc, false, false);
  }

  const float bv = bias ? bias[col0 + m] : 0.0f;
#pragma unroll
  for (int v = 0; v < 8; ++v) {
    const int r = row0 + v + (half << 3);
    out[(size_t)r * outC + col0 + m] = acc[v] + bv;
  }
}

// ---------------------------------------------------------------------------
// Degree / norm kernels
// ---------------------------------------------------------------------------
__global__ void fill_f32(float* __restrict__ p, float v, int n) {
  int i = blockIdx.x * blockDim.x + threadIdx.x;
  if (i < n) p[i] = v;
}

__global__ void deg_edges(const int* __restrict__ col, float* __restrict__ deg,
                          int nEdges) {
  int e = blockIdx.x * blockDim.x + threadIdx.x;
  if (e < nEdges) atomicAdd(&deg[col[e]], 1.0f);
}

__global__ void rsqrt_inplace(float* __restrict__ p, int n) {
  int i = blockIdx.x * blockDim.x + threadIdx.x;
  if (i < n) p[i] = rsqrtf(p[i]);
}

// ---------------------------------------------------------------------------
// agg[i,:] = h[i,:] * dinv[i]^2   (self-loop contribution, also initializes agg)
// ---------------------------------------------------------------------------
__global__ void agg_init(const float* __restrict__ H,
                         const float* __restrict__ dinv,
                         float* __restrict__ agg, int nElems) {
  int idx = blockIdx.x * blockDim.x + threadIdx.x;
  if (idx >= nElems) return;
  const float dv = dinv[idx >> 5];
  agg[idx] = H[idx] * dv * dv;
}

// ---------------------------------------------------------------------------
// Edge scatter: agg[c,:] += h[r,:] * dinv[r] * dinv[c]
// 8 threads per edge, float4 per thread; atomics land in L2 (12.8MB target).
// ---------------------------------------------------------------------------
__global__ void edge_scatter(const float* __restrict__ H,
                             const float* __restrict__ dinv,
                             const int* __restrict__ rowI,
                             const int* __restrict__ colI,
                             float* __restrict__ agg, int nEdges) {
  int tid = blockIdx.x * blockDim.x + threadIdx.x;
  int e = tid >> 3;
  if (e >= nEdges) return;
  int part = tid & 7;
  int r = rowI[e];
  int c = colI[e];
  float nrm = dinv[r] * dinv[c];
  const float4 hv = *(const float4*)(H + (size_t)r * DFEAT + part * 4);
  float* dst = agg + (size_t)c * DFEAT + part * 4;
  atomicAdd(dst + 0, hv.x * nrm);
  atomicAdd(dst + 1, hv.y * nrm);
  atomicAdd(dst + 2, hv.z * nrm);
  atomicAdd(dst + 3, hv.w * nrm);
}

// ---------------------------------------------------------------------------
// Epilogues
// ---------------------------------------------------------------------------
__global__ void bias_relu_inplace(float* __restrict__ Q,
                                  const float* __restrict__ b, int nElems) {
  int idx = blockIdx.x * blockDim.x + threadIdx.x;
  if (idx < nElems) Q[idx] = fmaxf(Q[idx] + b[idx & 31], 0.0f);
}

__global__ void res_bias_relu(const float* __restrict__ Q,
                              const float* __restrict__ b,
                              const float* __restrict__ xres,
                              float* __restrict__ xout, int nElems) {
  int idx = blockIdx.x * blockDim.x + threadIdx.x;
  if (idx < nElems) xout[idx] = fmaxf(Q[idx] + b[idx & 31] + xres[idx], 0.0f);
}

// ---------------------------------------------------------------------------
// Global mean pool
// ---------------------------------------------------------------------------
__global__ void pool_scatter(const float* __restrict__ X,
                             const int* __restrict__ batch,
                             float* __restrict__ sums,
                             float* __restrict__ cnt, int nNodes) {
  int tid = blockIdx.x * blockDim.x + threadIdx.x;
  int i = tid >> 3;
  if (i >= nNodes) return;
  int part = tid & 7;
  int g = batch[i];
  const float4 v = *(const float4*)(X + (size_t)i * DFEAT + part * 4);
  float* dst = sums + (size_t)g * DFEAT + part * 4;
  atomicAdd(dst + 0, v.x);
  atomicAdd(dst + 1, v.y);
  atomicAdd(dst + 2, v.z);
  atomicAdd(dst + 3, v.w);
  if (part == 0) atomicAdd(&cnt[g], 1.0f);
}

__global__ void pool_div(float* __restrict__ sums,
                         const float* __restrict__ cnt, int nElems) {
  int idx = blockIdx.x * blockDim.x + threadIdx.x;
  if (idx < nElems) sums[idx] /= fmaxf(cnt[idx >> 5], 1.0f);
}

// ---------------------------------------------------------------------------

static inline int gdim(long long n, int b) { return (int)((n + b - 1) / b); }

extern "C" void kernel_launch(void* const* d_in, const int* in_sizes, int n_in,
                              void* d_out, int out_size, void* d_ws,
                              size_t ws_size, hipStream_t stream) {
  const float* x     = (const float*)d_in[0];
  const int*   ei    = (const int*)d_in[1];
  const int*   batch = (const int*)d_in[2];
  const float* Wc[3][2] = {{(const float*)d_in[3],  (const float*)d_in[5]},
                           {(const float*)d_in[7],  (const float*)d_in[9]},
                           {(const float*)d_in[11], (const float*)d_in[13]}};
  const float* Bc[3][2] = {{(const float*)d_in[4],  (const float*)d_in[6]},
                           {(const float*)d_in[8],  (const float*)d_in[10]},
                           {(const float*)d_in[12], (const float*)d_in[14]}};
  const float* Wlin = (const float*)d_in[15];
  const float* blin = (const float*)d_in[16];

  const int N = in_sizes[0] / DFEAT;       // 100000 nodes
  const int E = in_sizes[1] / 2;           // 3.2M edges
  const int OUTC = in_sizes[16];           // 128
  const int G = out_size / OUTC;           // 8192 graphs
  const int* rowI = ei;
  const int* colI = ei + E;

  // Workspace layout (floats)
  float* ws   = (float*)d_ws;
  float* dinv = ws;                         // N
  float* P    = dinv + N;                   // N*32  (GEMM output h)
  float* Q    = P + (size_t)N * DFEAT;      // N*32  (aggregate)
  float* xc   = Q + (size_t)N * DFEAT;      // N*32  (residual stream)
  float* psum = xc + (size_t)N * DFEAT;     // G*32
  float* pcnt = psum + (size_t)G * DFEAT;   // G

  const int B256 = 256;
  const int NE = N * DFEAT;

  // --- symmetric normalization: dinv = rsqrt(in_degree + 1) ---
  fill_f32<<<gdim(N, B256), B256, 0, stream>>>(dinv, 1.0f, N);
  deg_edges<<<gdim(E, B256), B256, 0, stream>>>(colI, dinv, E);
  rsqrt_inplace<<<gdim(N, B256), B256, 0, stream>>>(dinv, N);

  // --- 3 residual GCN blocks ---
  const int gemmBlocks = gdim((long long)(N / 16) * (DFEAT / 16) * 32, B256);
  const float* xin = x;
  for (int l = 0; l < 3; ++l) {
    // conv1: h = xin @ W1 ; agg = norm-weighted sum ; t = relu(agg + b1)
    gemm_k32_wmma<<<gemmBlocks, B256, 0, stream>>>(xin, Wc[l][0], nullptr, P,
                                                   N, DFEAT);
    agg_init<<<gdim(NE, B256), B256, 0, stream>>>(P, dinv, Q, NE);
    edge_scatter<<<gdim((long long)E * 8, B256), B256, 0, stream>>>(
        P, dinv, rowI, colI, Q, E);
    bias_relu_inplace<<<gdim(NE, B256), B256, 0, stream>>>(Q, Bc[l][0], NE);

    // conv2: h = t @ W2 ; agg ; x = relu(agg + b2 + xin)
    gemm_k32_wmma<<<gemmBlocks, B256, 0, stream>>>(Q, Wc[l][1], nullptr, P,
                                                   N, DFEAT);
    agg_init<<<gdim(NE, B256), B256, 0, stream>>>(P, dinv, Q, NE);
    edge_scatter<<<gdim((long long)E * 8, B256), B256, 0, stream>>>(
        P, dinv, rowI, colI, Q, E);
    res_bias_relu<<<gdim(NE, B256), B256, 0, stream>>>(Q, Bc[l][1], xin, xc,
                                                       NE);
    xin = xc;
  }

  // --- global mean pool + final linear head (WMMA, bias fused) ---
  fill_f32<<<gdim((long long)G * DFEAT + G, B256), B256, 0, stream>>>(
      psum, 0.0f, G * DFEAT + G);  // psum and pcnt are contiguous
  pool_scatter<<<gdim((long long)N * 8, B256), B256, 0, stream>>>(
      xc, batch, psum, pcnt, N);
  pool_div<<<gdim((long long)G * DFEAT, B256), B256, 0, stream>>>(psum, pcnt,
                                                                  G * DFEAT);
  const int headBlocks = gdim((long long)(G / 16) * (OUTC / 16) * 32, B256);
  gemm_k32_wmma<<<headBlocks, B256, 0, stream>>>(psum, Wlin, blin,
                                                 (float*)d_out, G, OUTC);
}